// EncoderPositional_79242146612026
// MI455X (gfx1250) — compile-verified
//
#include <hip/hip_runtime.h>
#include <hip/hip_bf16.h>

typedef __attribute__((ext_vector_type(2))) float v2f;
typedef __attribute__((ext_vector_type(8))) float v8f;

#define VOCAB 50257
#define WE    512
#define PE    512
#define EMB   1024   // WE + PE
#define SEQ   4096

// ---------------------------------------------------------------------------
// Kernel 1: fused gather + concat.
// One block per row, 256 threads, one float4 (16B, GLOBAL_LOAD/STORE_B128)
// per thread. Threads 0..127 move the gathered word-embedding row (512 f32),
// threads 128..255 move the positional row (512 f32). idx[row] is a uniform
// (scalar) load per block. Fully coalesced: 23.3 TB/s-limited.
// ---------------------------------------------------------------------------
__global__ void __launch_bounds__(256)
gather_concat_kernel(const int* __restrict__ idx,
                     const float* __restrict__ word_table,
                     const float* __restrict__ pos_table,
                     float* __restrict__ out) {
  const int row = blockIdx.x;
  const int t   = threadIdx.x;

  const int w = idx[row];  // uniform per block -> s_load path

  const float4* __restrict__ wsrc =
      reinterpret_cast<const float4*>(word_table + (size_t)w * WE);
  const float4* __restrict__ psrc =
      reinterpret_cast<const float4*>(pos_table + (size_t)row * PE);
  float4* __restrict__ dst =
      reinterpret_cast<float4*>(out + (size_t)row * EMB);

  if (t < 128) {
    dst[t] = wsrc[t];            // cols [0, 512)
  } else {
    dst[t] = psrc[t - 128];      // cols [512, 1024)
  }
}

// ---------------------------------------------------------------------------
// Kernel 2: column mean via V_WMMA_F32_16X16X4_F32.
// A = 16x4 all-ones  =>  D[m][n] = sum_k B[k][n] + C[m][n]:
// each WMMA reduces a 4-row x 16-col f32 tile into the accumulator.
// Grid: 64 blocks (one per 16-column group) x 128 threads (4 waves).
// Each wave reduces a 1024-row slab (256 chained WMMAs); the 4 per-wave
// partials are combined through LDS by wave 0 and scaled by 1/4096.
// B-operand load: lane l supplies (k = half + {0,2}, n = l%16); with a ones
// A-matrix the k->lane mapping is sum-invariant, n matches the C/D layout.
// EXEC is all-ones for every WMMA (no divergence before the final store).
// ---------------------------------------------------------------------------
__global__ void __launch_bounds__(128)
col_mean_wmma_kernel(const float* __restrict__ out,
                     float* __restrict__ hidden) {
  const int lane    = threadIdx.x & 31;
  const int waveId  = threadIdx.x >> 5;       // 0..3
  const int n       = lane & 15;              // column within the 16-group
  const int half    = (lane >> 4) & 1;        // lane group selects K row pair
  const int colBase = blockIdx.x * 16;

  const int rowsPerWave = SEQ / 4;            // 1024
  const int rowStart    = waveId * rowsPerWave;

  v2f a; a.x = 1.0f; a.y = 1.0f;              // 16x4 ones matrix
  v8f acc = {};

  const float* __restrict__ base = out + colBase + n;
  #pragma unroll 4
  for (int r = rowStart; r < rowStart + rowsPerWave; r += 4) {
    v2f b;
    b.x = base[(size_t)(r + half) * EMB];
    b.y = base[(size_t)(r + 2 + half) * EMB];
    // (neg_a, A, neg_b, B, c_mod, C, reuse_a, reuse_b)
    acc = __builtin_amdgcn_wmma_f32_16x16x4_f32(
        false, a, false, b, (short)0, acc, false, false);
  }

  // All 16 rows of D are identical column sums; row M=0 lives in acc[0],
  // lanes 0..15 (N = lane). Combine the 4 wave partials through LDS.
  __shared__ float partial[4 * 16];
  if (lane < 16) partial[waveId * 16 + n] = acc[0];
  __syncthreads();

  if (threadIdx.x < 16) {
    const int c = threadIdx.x;
    float s = partial[c] + partial[16 + c] + partial[32 + c] + partial[48 + c];
    hidden[colBase + c] = s * (1.0f / (float)SEQ);
  }
}

// ---------------------------------------------------------------------------
// Launch: d_in = {inputs(int), word_table(f32), pos_table(f32), max_length}.
// d_out = output (4096*1024 f32) ++ hidden (1024 f32).
// ---------------------------------------------------------------------------
extern "C" void kernel_launch(void* const* d_in, const int* in_sizes, int n_in,
                              void* d_out, int out_size, void* d_ws, size_t ws_size,
                              hipStream_t stream) {
  (void)in_sizes; (void)n_in; (void)out_size; (void)d_ws; (void)ws_size;

  const int*   idx        = (const int*)d_in[0];
  const float* word_table = (const float*)d_in[1];
  const float* pos_table  = (const float*)d_in[2];

  float* out    = (float*)d_out;
  float* hidden = out + (size_t)SEQ * EMB;

  gather_concat_kernel<<<SEQ, 256, 0, stream>>>(idx, word_table, pos_table, out);
  col_mean_wmma_kernel<<<EMB / 16, 128, 0, stream>>>(out, hidden);
}